// disparity_attention2_61443802137023
// MI455X (gfx1250) — compile-verified
//
#include <hip/hip_runtime.h>

// ---- problem constants ----
#define Bn 4
#define Cc 32
#define Dd 48
#define Hh 64
#define Ww 128
#define HW (Hh*Ww)          // 8192
#define CIn 16
#define CRn 8
#define XSTR_C (Dd*HW)      // 393216  (stride between channels, elems)
#define XSTR_B (Cc*Dd*HW)   // 12582912 (stride between batches, elems)
#define BN_EPS 1e-5f
#define XT (Dd*16)          // 768: LDS x-tile [c][d][hwl] inner extent

typedef __attribute__((ext_vector_type(2))) float v2f;
typedef __attribute__((ext_vector_type(8))) float v8f;
typedef __attribute__((ext_vector_type(4))) unsigned int v4u;
typedef __attribute__((ext_vector_type(8))) int v8i;
typedef __attribute__((ext_vector_type(4))) int v4i;

// ---------------------------------------------------------------------
// TDM: async DMA of a 3-D fp32 tile (dim0 contiguous) from global -> LDS.
// D# packed per CDNA5 ISA 8.3-8.6. Wave-level op: caller guards to wave 0.
// ---------------------------------------------------------------------
__device__ __forceinline__ void tdm_load_3d(
    unsigned lds_addr, unsigned long long gaddr,
    unsigned dim0, unsigned dim1, unsigned dim2,            // tensor extents (elems)
    unsigned long long s0, unsigned long long s1,           // line strides (elems)
    unsigned t0, unsigned t1, unsigned t2)                  // tile dims
{
    v4u g0;
    g0[0] = 1u;                                             // count=1, user mode
    g0[1] = lds_addr;                                       // LDS byte offset
    g0[2] = (unsigned)(gaddr & 0xffffffffull);              // global_addr[31:0]
    g0[3] = (unsigned)((gaddr >> 32) & 0x1ffffffull)        // global_addr[56:32]
          | (2u << 30);                                     // type=2 (image)
    v8i g1;
    g1[0] = (int)(2u << 16);                                // data_size=4B, mask=0
    g1[1] = (int)(dim0 << 16);                              // tensor_dim0[15:0]
    g1[2] = (int)((dim0 >> 16) | (dim1 << 16));             // dim0 hi | dim1 lo
    g1[3] = (int)((dim1 >> 16) | (t0 << 16));               // dim1 hi | tile_dim0
    g1[4] = (int)(t1 | (t2 << 16));                         // tile_dim1 | tile_dim2
    g1[5] = (int)(s0 & 0xffffffffull);                      // dim0_stride[31:0]
    g1[6] = (int)(((unsigned)(s0 >> 32) & 0xffffu)          // dim0_stride[47:32]
          | ((unsigned)(s1 & 0xffffull) << 16));            // dim1_stride[15:0]
    g1[7] = (int)((s1 >> 16) & 0xffffffffull);              // dim1_stride[47:16]
    v4i g2;
    g2[0] = (int)dim2;                                      // tensor_dim2
    g2[1] = 0;                                              // tensor_dim3
    g2[2] = 0;                                              // tensor_dim2_stride lo
    g2[3] = 0;                                              // stride hi | tile_dim3=0
    v4i g3 = {0, 0, 0, 0};
#if defined(__clang_major__) && (__clang_major__ >= 23)
    v8i g4 = {0, 0, 0, 0, 0, 0, 0, 0};
    __builtin_amdgcn_tensor_load_to_lds(g0, g1, g2, g3, g4, 0);
#else
    __builtin_amdgcn_tensor_load_to_lds(g0, g1, g2, g3, 0);
#endif
}

__device__ __forceinline__ unsigned lds_off(const void* p)
{   // generic LDS pointer: [31:0] is the workgroup-relative LDS byte offset
    return (unsigned)(unsigned long long)(uintptr_t)p;
}

// =====================================================================
// Kernel 1: TDM-fed fused Q/K projection + energy gram partials (fp32 WMMA).
// Grid: 256 WGs (b, 64 hw-chunks of 128), 288 threads (9 waves).
// Per 16-hw tile: TDM prefetch of next x tile overlaps the WMMA stage.
// =====================================================================
__global__ __launch_bounds__(288) void k_energy_partial(
    const float* __restrict__ x, const float* __restrict__ wq,
    const float* __restrict__ wk, float* __restrict__ partial)
{
    __shared__ float Xs[Cc * XT];   // x tile [c][d][hwl], TDM-sequential layout
    __shared__ float Qs[Dd][260];   // [d][k], k = o*16+hwl, padded for banks
    __shared__ float Ks[Dd][260];
    __shared__ float wqs[CIn][Cc];
    __shared__ float wks[CIn][Cc];

    const int tid    = threadIdx.x;
    const int wg     = blockIdx.x;       // 0..255
    const int b      = wg >> 6;
    const int chunk  = wg & 63;
    const int hwbase = chunk * 128;      // 8 tiles of 16

    const unsigned xs_lds = lds_off(&Xs[0]);
    const unsigned long long xbase =
        (unsigned long long)(uintptr_t)x + ((unsigned long long)b * XSTR_B) * 4ull;

    const int lane = tid & 31;
    const int wave = tid >> 5;                 // 0..8
    if (wave == 0)                              // prefetch tile 0
        tdm_load_3d(xs_lds, xbase + (unsigned long long)hwbase * 4ull,
                    HW, Dd, Cc, HW, XSTR_C, 16, Dd, Cc);

    for (int i = tid; i < CIn * Cc; i += 288) { wqs[0][i] = wq[i]; wks[0][i] = wk[i]; }

    const int dt   = (wave / 3) * 16;
    const int et   = (wave % 3) * 16;
    const int m    = lane & 15;
    const int koff = (lane >> 4) << 1;         // 0 or 2

    v8f acc = {};

    for (int t = 0; t < 8; ++t) {
        if (wave == 0) __builtin_amdgcn_s_wait_tensorcnt(0);
        __syncthreads();                       // Xs(t) ready; prev WMMA done
        // ---- stage 1: Q/K projection from the LDS x tile ----
        for (int i = 0; i < 3; ++i) {
            int idx = tid + i * 288;           // (d, hwl) over 768
            if (idx < XT) {
                int d = idx >> 4, hwl = idx & 15;
                float q[CIn], kk[CIn];
                #pragma unroll
                for (int o = 0; o < CIn; ++o) { q[o] = 0.f; kk[o] = 0.f; }
                for (int c0 = 0; c0 < Cc; c0 += 8) {
                    float xc[8];
                    #pragma unroll
                    for (int c = 0; c < 8; ++c) xc[c] = Xs[(c0 + c) * XT + idx];
                    #pragma unroll
                    for (int o = 0; o < CIn; ++o) {
                        float qq = q[o], kx = kk[o];
                        #pragma unroll
                        for (int c = 0; c < 8; ++c) {
                            qq += wqs[o][c0 + c] * xc[c];
                            kx += wks[o][c0 + c] * xc[c];
                        }
                        q[o] = qq; kk[o] = kx;
                    }
                }
                #pragma unroll
                for (int o = 0; o < CIn; ++o) {
                    Qs[d][o * 16 + hwl] = q[o];
                    Ks[d][o * 16 + hwl] = kk[o];
                }
            }
        }
        __syncthreads();                       // Qs/Ks ready; Xs fully consumed
        if (wave == 0 && t < 7)                // prefetch next tile under WMMA
            tdm_load_3d(xs_lds,
                        xbase + (unsigned long long)(hwbase + (t + 1) * 16) * 4ull,
                        HW, Dd, Cc, HW, XSTR_C, 16, Dd, Cc);
        // ---- stage 2: E[dt:+16, et:+16] += Q_tile * K_tile^T  (k=256) ----
        for (int k0 = 0; k0 < 256; k0 += 4) {
            v2f a  = *(const v2f*)&Qs[dt + m][k0 + koff];
            v2f bb = *(const v2f*)&Ks[et + m][k0 + koff];
            acc = __builtin_amdgcn_wmma_f32_16x16x4_f32(
                false, a, false, bb, (short)0, acc, false, false);
        }
        __syncthreads();
    }

    float* P = partial + (size_t)wg * (Dd * Dd);
    #pragma unroll
    for (int r = 0; r < 8; ++r) {
        int row = dt + ((lane < 16) ? r : r + 8);
        P[row * Dd + (et + m)] = acc[r];
    }
}

// =====================================================================
// Kernel 2: deterministic fixed-order reduction of 64 chunk-partials per b.
// =====================================================================
__global__ __launch_bounds__(256) void k_reduce_energy(
    const float* __restrict__ partial, float* __restrict__ energy)
{
    int idx = blockIdx.x * 256 + threadIdx.x;     // 0..9215
    if (idx >= Bn * Dd * Dd) return;
    int b = idx / (Dd * Dd), de = idx % (Dd * Dd);
    float s = 0.f;
    for (int j = 0; j < 64; ++j)
        s += partial[((size_t)(b * 64 + j)) * (Dd * Dd) + de];
    energy[idx] = s;
}

// =====================================================================
// Kernel 3: global min/max normalize + row softmax (single workgroup).
// =====================================================================
__global__ __launch_bounds__(256) void k_softmax(
    const float* __restrict__ energy, float* __restrict__ attn)
{
    __shared__ float smin[256], smax[256];
    int tid = threadIdx.x;
    float mn = 3.4e38f, mx = -3.4e38f;
    for (int i = tid; i < Bn * Dd * Dd; i += 256) {
        float v = energy[i]; mn = fminf(mn, v); mx = fmaxf(mx, v);
    }
    smin[tid] = mn; smax[tid] = mx;
    __syncthreads();
    for (int s = 128; s > 0; s >>= 1) {
        if (tid < s) {
            smin[tid] = fminf(smin[tid], smin[tid + s]);
            smax[tid] = fmaxf(smax[tid], smax[tid + s]);
        }
        __syncthreads();
    }
    float gmin = smin[0];
    float inv  = 1.0f / (smax[0] - gmin);
    if (tid < Bn * Dd) {
        const float* row = energy + tid * Dd;
        float e[Dd];
        float rmax = -3.4e38f;
        for (int j = 0; j < Dd; ++j) { e[j] = (row[j] - gmin) * inv; rmax = fmaxf(rmax, e[j]); }
        float s = 0.f;
        for (int j = 0; j < Dd; ++j) { e[j] = __expf(e[j] - rmax); s += e[j]; }
        float r = 1.0f / s;
        for (int j = 0; j < Dd; ++j) attn[tid * Dd + j] = e[j] * r;
    }
}

// =====================================================================
// Kernel 4: TDM x-tile + fused V projection + (attn x V) WMMA +
// conv_up/BN/ReLU/conv_up2 + residual (residual read from LDS tile).
// Grid: B*512 WGs (one 16-wide hw tile), 256 threads (8 waves).
// =====================================================================
__global__ __launch_bounds__(256) void k_attend_out(
    const float* __restrict__ x,  const float* __restrict__ wv,
    const float* __restrict__ attn,
    const float* __restrict__ w1, const float* __restrict__ b1,
    const float* __restrict__ bng, const float* __restrict__ bnb,
    const float* __restrict__ bnm, const float* __restrict__ bnv,
    const float* __restrict__ w2, const float* __restrict__ b2,
    const float* __restrict__ gamma, float* __restrict__ out)
{
    __shared__ float Xs[Cc * XT];    // x tile [c][e][hwl]
    __shared__ float As[Dd][52];     // attn [d][e], padded
    __shared__ float Vs[256][52];    // V operand [n][e]  (n = o*16+hwl)
    __shared__ float Os[Dd][256];    // out [d][n]
    __shared__ float wvs[CIn][Cc];
    __shared__ float w1s[CRn][CIn], w2s[Cc][CRn];
    __shared__ float b1s[CRn], invs[CRn], shfs[CRn], b2s[Cc], gma;

    const int tid = threadIdx.x;
    const int wg  = blockIdx.x;
    const int b   = wg >> 9;
    const int hw0 = (wg & 511) * 16;
    const int lane = tid & 31, wave = tid >> 5;

    if (wave == 0)                               // async x tile DMA up front
        tdm_load_3d(lds_off(&Xs[0]),
                    (unsigned long long)(uintptr_t)x +
                        ((unsigned long long)b * XSTR_B + (unsigned)hw0) * 4ull,
                    HW, Dd, Cc, HW, XSTR_C, 16, Dd, Cc);

    for (int i = tid; i < CIn*Cc; i += 256) wvs[0][i] = wv[i];
    for (int i = tid; i < CRn*CIn; i += 256) w1s[0][i] = w1[i];
    for (int i = tid; i < Cc*CRn; i += 256) w2s[0][i] = w2[i];
    if (tid < CRn) {
        float iv = bng[tid] * rsqrtf(bnv[tid] + BN_EPS);
        invs[tid] = iv; shfs[tid] = bnb[tid] - bnm[tid] * iv; b1s[tid] = b1[tid];
    }
    if (tid < Cc) b2s[tid] = b2[tid];
    if (tid == 0) gma = gamma[0];
    for (int i = tid; i < Dd * Dd; i += 256) As[i / Dd][i % Dd] = attn[b * Dd * Dd + i];
    if (wave == 0) __builtin_amdgcn_s_wait_tensorcnt(0);
    __syncthreads();

    // ---- V projection from the LDS x tile ----
    for (int i = 0; i < 3; ++i) {
        int idx = tid + i * 256;                 // (e, hwl) over 768 exactly
        int hwl = idx & 15;
        float vv[CIn];
        #pragma unroll
        for (int o = 0; o < CIn; ++o) vv[o] = 0.f;
        for (int c0 = 0; c0 < Cc; c0 += 8) {
            float xc[8];
            #pragma unroll
            for (int c = 0; c < 8; ++c) xc[c] = Xs[(c0 + c) * XT + idx];
            #pragma unroll
            for (int o = 0; o < CIn; ++o) {
                float a = vv[o];
                #pragma unroll
                for (int c = 0; c < 8; ++c) a += wvs[o][c0 + c] * xc[c];
                vv[o] = a;
            }
        }
        #pragma unroll
        for (int o = 0; o < CIn; ++o) Vs[o * 16 + hwl][idx >> 4] = vv[o];
    }
    __syncthreads();

    // ---- out[48][256] = attn[48][48] x V[48][256] via WMMA, k=48 ----
    const int m = lane & 15, koff = (lane >> 4) << 1;
    for (int t = 0; t < 6; ++t) {
        int tl = wave * 6 + t;                   // 48 tiles, 6 per wave
        int dt = (tl >> 4) * 16, nt = (tl & 15) * 16;
        v8f acc = {};
        #pragma unroll
        for (int k0 = 0; k0 < Dd; k0 += 4) {
            v2f a  = *(const v2f*)&As[dt + m][k0 + koff];
            v2f bb = *(const v2f*)&Vs[nt + m][k0 + koff];
            acc = __builtin_amdgcn_wmma_f32_16x16x4_f32(
                false, a, false, bb, (short)0, acc, false, false);
        }
        #pragma unroll
        for (int r = 0; r < 8; ++r) {
            int row = dt + ((lane < 16) ? r : r + 8);
            Os[row][nt + m] = acc[r];
        }
    }
    __syncthreads();

    // ---- per-voxel conv_up -> BN -> ReLU -> conv_up2 -> residual(LDS) ----
    for (int i = 0; i < 3; ++i) {
        int idx = tid + i * 256;                 // (d, hwl) over 768 exactly
        int d = idx >> 4, hwl = idx & 15;
        float o16[CIn];
        #pragma unroll
        for (int o = 0; o < CIn; ++o) o16[o] = Os[d][o * 16 + hwl];
        float y1[CRn];
        #pragma unroll
        for (int r = 0; r < CRn; ++r) {
            float y = b1s[r];
            #pragma unroll
            for (int o = 0; o < CIn; ++o) y += w1s[r][o] * o16[o];
            y1[r] = fmaxf(y * invs[r] + shfs[r], 0.f);
        }
        float* op = out + (size_t)b * XSTR_B + (size_t)d * HW + hw0 + hwl;
        #pragma unroll
        for (int c = 0; c < Cc; ++c) {
            float y = b2s[c];
            #pragma unroll
            for (int r = 0; r < CRn; ++r) y += w2s[c][r] * y1[r];
            op[(size_t)c * XSTR_C] = gma * y + Xs[c * XT + idx];
        }
    }
}

// =====================================================================
extern "C" void kernel_launch(void* const* d_in, const int* in_sizes, int n_in,
                              void* d_out, int out_size, void* d_ws, size_t ws_size,
                              hipStream_t stream)
{
    const float* x        = (const float*)d_in[0];
    const float* wq       = (const float*)d_in[1];
    const float* wk       = (const float*)d_in[2];
    const float* wv       = (const float*)d_in[3];
    const float* w_up1    = (const float*)d_in[4];
    const float* b_up1    = (const float*)d_in[5];
    const float* bn_gamma = (const float*)d_in[6];
    const float* bn_beta  = (const float*)d_in[7];
    const float* bn_mean  = (const float*)d_in[8];
    const float* bn_var   = (const float*)d_in[9];
    const float* w_up2    = (const float*)d_in[10];
    const float* b_up2    = (const float*)d_in[11];
    const float* gamma    = (const float*)d_in[12];
    float* out = (float*)d_out;

    float* partial = (float*)d_ws;                     // 256 * 2304 floats (2.4 MB)
    float* energy  = partial + 256 * Dd * Dd;          // 9216 floats
    float* attn    = energy  + Bn * Dd * Dd;           // 9216 floats

    k_energy_partial<<<256, 288, 0, stream>>>(x, wq, wk, partial);
    k_reduce_energy<<<36, 256, 0, stream>>>(partial, energy);
    k_softmax<<<1, 256, 0, stream>>>(energy, attn);
    k_attend_out<<<Bn * 512, 256, 0, stream>>>(x, wv, attn, w_up1, b_up1,
        bn_gamma, bn_beta, bn_mean, bn_var, w_up2, b_up2, gamma, out);
}